// DotProductAttention_25091198943263
// MI455X (gfx1250) — compile-verified
//
#include <hip/hip_runtime.h>

typedef float v2f __attribute__((ext_vector_type(2)));
typedef float v4f __attribute__((ext_vector_type(4)));
typedef float v8f __attribute__((ext_vector_type(8)));

#define NROWS 8192
#define DDIM  1024

// ---------------------------------------------------------------------------
// Pass 1: row-wise dot products via V_WMMA_F32_16X16X4_F32 (exact f32 math).
// One wave32 handles a 16-row tile. A = Q[16 x 4], B = K^T[4 x 16]; the
// accumulator diagonal after 256 k-steps is raw[m] = <q_m, k_m>.
//
// A layout (32-bit 16x4): lane L<16 holds row M=L, K={k,k+1} in VGPR0/1;
// lane L>=16 holds row M=L-16, K={k+2,k+3}. B (4x16) mirrors this with
// column N in place of row M, so both operands load with the same pattern.
// ---------------------------------------------------------------------------
__global__ __launch_bounds__(256) void rowdot_wmma(const float* __restrict__ q,
                                                   const float* __restrict__ k,
                                                   float* __restrict__ raw) {
    const int lane = threadIdx.x & 31;
    const int wave = threadIdx.x >> 5;
    const int tile = blockIdx.x * 8 + wave;          // 512 tiles of 16 rows
    const int row  = tile * 16 + (lane & 15);
    const int koff = (lane >> 4) * 2;                // lanes 0-15: k+0,1 ; 16-31: k+2,3

    const float* __restrict__ qp = q + (size_t)row * DDIM + koff;
    const float* __restrict__ kp = k + (size_t)row * DDIM + koff;

    v8f c = {0.f, 0.f, 0.f, 0.f, 0.f, 0.f, 0.f, 0.f};
#pragma unroll 4
    for (int kk = 0; kk < DDIM; kk += 4) {
        v2f a = *(const v2f*)(qp + kk);
        v2f b = *(const v2f*)(kp + kk);
        // 8 args: (neg_a, A, neg_b, B, c_mod, C, reuse_a, reuse_b)
        c = __builtin_amdgcn_wmma_f32_16x16x4_f32(false, a, false, b,
                                                  (short)0, c, false, false);
    }

    // Diagonal extraction from the 16x16 f32 C layout:
    //   m in 0..7  -> VGPR m,   lane m        (lanes 0..7)
    //   m in 8..15 -> VGPR m-8, lane m+16     (lanes 24..31)
    float vals[8];
#pragma unroll
    for (int i = 0; i < 8; ++i) vals[i] = c[i];

    if (lane < 8) {
        raw[tile * 16 + lane] = vals[lane];
    } else if (lane >= 24) {
        raw[tile * 16 + (lane - 16)] = vals[lane - 24];
    }
}

// ---------------------------------------------------------------------------
// Pass 2: single-block numerically-stable softmax over 8192 logits.
// 1024 threads x 8 elements each. Writes normalized probabilities to sm[].
// ---------------------------------------------------------------------------
__global__ __launch_bounds__(1024) void softmax_8192(const float* __restrict__ raw,
                                                     float* __restrict__ sm) {
    __shared__ float red[1024];
    const int t = threadIdx.x;

    float v[8];
    float lmax = -3.402823466e+38f;
#pragma unroll
    for (int j = 0; j < 8; ++j) {
        v[j] = raw[t + 1024 * j];
        lmax = fmaxf(lmax, v[j]);
    }
    red[t] = lmax;
    __syncthreads();
    for (int s = 512; s > 0; s >>= 1) {
        if (t < s) red[t] = fmaxf(red[t], red[t + s]);
        __syncthreads();
    }
    const float gmax = red[0];
    __syncthreads();

    float lsum = 0.f;
#pragma unroll
    for (int j = 0; j < 8; ++j) {
        v[j] = expf(v[j] - gmax);
        lsum += v[j];
    }
    red[t] = lsum;
    __syncthreads();
    for (int s = 512; s > 0; s >>= 1) {
        if (t < s) red[t] += red[t + s];
        __syncthreads();
    }
    const float inv = 1.0f / red[0];

#pragma unroll
    for (int j = 0; j < 8; ++j) sm[t + 1024 * j] = v[j] * inv;
}

// ---------------------------------------------------------------------------
// Pass 3 (dominant cost, ~256 MB): stream zeros with 128-bit non-temporal
// stores, splicing the diagonal probability into the covering float4.
// Output > L2 (256 MB vs 192 MB) and write-once, so NT avoids L2 thrash.
// ---------------------------------------------------------------------------
__global__ __launch_bounds__(256) void fill_diag(const float* __restrict__ sm,
                                                 float* __restrict__ out) {
    const size_t t    = (size_t)blockIdx.x * 256 + threadIdx.x;  // one float4 each
    const size_t base = t * 4;
    const int row  = (int)(base >> 13);            // NROWS = 8192 = 2^13
    const int col0 = (int)(base & (size_t)(NROWS - 1));

    float a0 = 0.f, a1 = 0.f, a2 = 0.f, a3 = 0.f;
    const int d = row - col0;                      // diag lands at col == row
    if (d >= 0 && d < 4) {
        const float s = sm[row];
        if (d == 0) a0 = s;
        else if (d == 1) a1 = s;
        else if (d == 2) a2 = s;
        else a3 = s;
    }
    v4f v = {a0, a1, a2, a3};
    __builtin_nontemporal_store(v, (v4f*)(out + base));
}

// ---------------------------------------------------------------------------
extern "C" void kernel_launch(void* const* d_in, const int* in_sizes, int n_in,
                              void* d_out, int out_size, void* d_ws, size_t ws_size,
                              hipStream_t stream) {
    (void)in_sizes; (void)n_in; (void)out_size; (void)ws_size;

    const float* q  = (const float*)d_in[0];
    const float* k  = (const float*)d_in[1];
    float*       out = (float*)d_out;

    float* raw = (float*)d_ws;       // 8192 floats
    float* sm  = raw + NROWS;        // 8192 floats

    // 512 wave-tiles of 16 rows -> 64 blocks x 8 waves
    rowdot_wmma<<<64, 256, 0, stream>>>(q, k, raw);
    softmax_8192<<<1, 1024, 0, stream>>>(raw, sm);
    // 8192*8192/4 float4s / 256 threads = 65536 blocks
    fill_diag<<<65536, 256, 0, stream>>>(sm, out);
}